// Encoder_57913339020010
// MI455X (gfx1250) — compile-verified
//
#include <hip/hip_runtime.h>
#include <math.h>

#define BB 4
#define LL 2048
#define DD 32

typedef __attribute__((ext_vector_type(2))) float f32x2;
typedef __attribute__((ext_vector_type(8))) float f32x8;

// ---------------------------------------------------------------------------
// Phase 1: per-position projections via V_WMMA_F32_16X16X4_F32.
//   E = type_table[event_type]  (row per position, 32 wide)
//   W columns: n=0: w_g[0:32], n=1: w_s[0:32], n=2: w_d[0:32],
//              n=3: w_g[32:64], n=4: w_s[32:64], n=5: w_d[32:64], n>=6: 0
//   P = E x W  -> per-position u_g,u_s,u_d (j-side), v_g,v_s,v_d (i-side).
// Epilogue stores exponentiated forms so the pairwise kernel needs only
// rcp/log/exp per pair:
//   proj[row*8+0] = exp(-(u_g + b_g))   (A_j)
//   proj[row*8+1] = exp(u_s + b_s)      (C_j)
//   proj[row*8+2] = exp(u_d + b_d)      (E_j)
//   proj[row*8+3] = exp(-v_g)           (B_i)
//   proj[row*8+4] = exp(v_s)            (D_i)
//   proj[row*8+5] = exp(v_d)            (F_i)
// One wave handles a 16-row tile; K=32 via 8 chained 16x16x4 WMMAs.
// Fragment build and epilogue are fully branchless (v_cndmask selects, no
// EXEC-mask diamonds); only the final 6-column store is lane-guarded.
// ---------------------------------------------------------------------------
__global__ __launch_bounds__(256) void proj_wmma_kernel(
    const int* __restrict__ et, const float* __restrict__ table,
    const float* __restrict__ wg, const float* __restrict__ wsv,
    const float* __restrict__ wd, const float* __restrict__ bg,
    const float* __restrict__ bs, const float* __restrict__ bd,
    float* __restrict__ proj, int nrows)
{
    int gtid = blockIdx.x * blockDim.x + threadIdx.x;
    int wave = gtid >> 5;
    int lane = threadIdx.x & 31;
    int row0 = wave * 16;
    if (row0 >= nrows) return;          // wave-uniform: EXEC stays all-ones

    int m  = lane & 15;                 // A: M row index; B/C: N col index
    int kh = lane >> 4;                 // K half selector (K and K+2 pattern)

    // --- A fragment source: one emb row per lane ---------------------------
    int typ = et[row0 + m];
    const float* erow = table + typ * DD;

    // --- branchless B-column blend masks -----------------------------------
    // col n: 0/3 -> w_g, 1/4 -> w_s, 2/5 -> w_d; offset 32 for n in [3,6); 0 else
    float c0 = (m == 0 || m == 3) ? 1.0f : 0.0f;
    float c1 = (m == 1 || m == 4) ? 1.0f : 0.0f;
    float c2 = (m == 2 || m == 5) ? 1.0f : 0.0f;
    int   off = (m >= 3 && m < 6) ? DD : 0;     // index stays in [0,63]: valid

    float av[16], bv[16];
#pragma unroll
    for (int s = 0; s < 8; ++s) {
        int ka = s * 4 + 2 * kh;        // K index for this lane half, this step
        av[2 * s]     = erow[ka];
        av[2 * s + 1] = erow[ka + 1];
        int i0 = off + ka, i1 = off + ka + 1;
        bv[2 * s]     = wg[i0] * c0 + wsv[i0] * c1 + wd[i0] * c2;
        bv[2 * s + 1] = wg[i1] * c0 + wsv[i1] * c1 + wd[i1] * c2;
    }

    f32x8 acc = (f32x8)0.0f;
#pragma unroll
    for (int s = 0; s < 8; ++s) {
        f32x2 a;  a[0] = av[2 * s];  a[1] = av[2 * s + 1];
        f32x2 bf; bf[0] = bv[2 * s]; bf[1] = bv[2 * s + 1];
        // (neg_a, A, neg_b, B, c_mod, C, reuse_a, reuse_b)
        acc = __builtin_amdgcn_wmma_f32_16x16x4_f32(
            false, a, false, bf, (short)0, acc, false, false);
    }

    // --- branchless epilogue: e = exp(sgn * (x + bias)) --------------------
    // m=0: -(x+b_g)  m=1: x+b_s  m=2: x+b_d  m=3: -x  m=4,5: x
    float bias = (m == 0) ? bg[0] : (m == 1) ? bs[0] : (m == 2) ? bd[0] : 0.0f;
    float sgn  = (m == 0 || m == 3) ? -1.0f : 1.0f;

    if (m < 6) {                        // lane-guarded store only
#pragma unroll
        for (int v = 0; v < 8; ++v) {
            int row = row0 + v + 8 * kh;        // C layout: M = v + 8*kh, N = m
            proj[row * 8 + m] = __expf(sgn * (acc[v] + bias));
        }
    }
}

// ---------------------------------------------------------------------------
// Phase 1b: hidden_vector [B,L,2D]: pe = [sin(arc+phi), cos(arc+phi)], emb.
// Trivially small (2 MB); one thread per element.
// ---------------------------------------------------------------------------
__global__ __launch_bounds__(256) void hidden_kernel(
    const float* __restrict__ time, const int* __restrict__ et,
    const float* __restrict__ Wt, const float* __restrict__ table,
    float* __restrict__ out)
{
    int idx = blockIdx.x * blockDim.x + threadIdx.x;   // pos*64 + k
    int k   = idx & 63;
    int pos = idx >> 6;
    int l   = pos & (LL - 1);
    float val;
    if (k < DD) {
        int h = k & 15;
        float phi = time[pos] * Wt[h];
        // div_term[h] = exp(2h * (-ln(10000)/32)) = exp(-h * ln(10000)/16)
        float div = __expf((float)h * -0.5756462732485114f);
        float arc = (float)l * div + phi;
        val = (k < 16) ? sinf(arc) : cosf(arc);
    } else {
        val = table[et[pos] * DD + (k - DD)];
    }
    out[idx] = val;
}

// ---------------------------------------------------------------------------
// Phase 2: pairwise scores + t_diff  (store-bandwidth bound: 136 MB out,
// ~5.8 us floor at 23.3 TB/s). Block = 256 threads covering an 8(i) x 128(j)
// tile; each thread owns 4 consecutive j -> one float4 store to scores and
// one to t_diff. Per pair: 1 rcp + 2 log + 1 exp (exp-forms precomputed).
// ---------------------------------------------------------------------------
__global__ __launch_bounds__(256) void pair_kernel(
    const float* __restrict__ time, const float* __restrict__ proj,
    float* __restrict__ scores, float* __restrict__ tdout)
{
    int b  = blockIdx.z;
    int i  = blockIdx.y * 8 + (threadIdx.x >> 5);
    int j0 = blockIdx.x * 128 + (threadIdx.x & 31) * 4;

    const float* tb = time + b * LL;
    const float* pb = proj + (size_t)b * LL * 8;

    float  ti  = tb[i];
    float4 ilo = *(const float4*)(pb + (size_t)i * 8);      // {A,C,E,B}
    float4 ihi = *(const float4*)(pb + (size_t)i * 8 + 4);  // {D,F,-,-}
    float  Bi = ilo.w, Di = ihi.x, Fi = ihi.y;

    float4 tj = *(const float4*)(tb + j0);
    const float* tjp = &tj.x;

    float out_s[4], out_t[4];
#pragma unroll
    for (int q = 0; q < 4; ++q) {
        int j = j0 + q;
        float4 jlo = *(const float4*)(pb + (size_t)j * 8);  // {A_j,C_j,E_j,B_j}
        float tdiff = fabsf(tjp[q] - ti);
        float gate  = 1.0f / (1.0f + jlo.x * Bi);           // sigmoid
        float scale = __logf(1.0f + jlo.y * Di);            // softplus
        float decay = __logf(1.0f + jlo.z * Fi);            // softplus
        float s = gate * scale * __expf(-decay * tdiff);
        out_t[q] = tdiff;
        out_s[q] = (i > j) ? s : 0.0f;                      // strict lower tri
    }

    size_t base = ((size_t)(b * LL + i)) * LL + j0;
    *(float4*)(scores + base) = make_float4(out_s[0], out_s[1], out_s[2], out_s[3]);
    *(float4*)(tdout  + base) = make_float4(out_t[0], out_t[1], out_t[2], out_t[3]);
}

// ---------------------------------------------------------------------------
extern "C" void kernel_launch(void* const* d_in, const int* in_sizes, int n_in,
                              void* d_out, int out_size, void* d_ws, size_t ws_size,
                              hipStream_t stream)
{
    const int*   et    = (const int*)  d_in[0];   // event_type  [B,L]
    const float* time  = (const float*)d_in[1];   // event_time  [B,L]
    const float* Wt    = (const float*)d_in[2];   // [16]
    const float* table = (const float*)d_in[3];   // [21,32]
    const float* wg    = (const float*)d_in[4];   // [64]
    const float* bg    = (const float*)d_in[5];   // [1]
    const float* wsv   = (const float*)d_in[6];   // [64]
    const float* bs    = (const float*)d_in[7];   // [1]
    const float* wd    = (const float*)d_in[8];   // [64]
    const float* bd    = (const float*)d_in[9];   // [1]

    float* out    = (float*)d_out;
    float* scores = out;                                   // [B,L,L]
    float* hidden = out + (size_t)BB * LL * LL;            // [B,L,2D]
    float* tdout  = hidden + (size_t)BB * LL * 2 * DD;     // [B,L,L]

    float* proj = (float*)d_ws;                            // [B*L][8] = 256 KB

    const int nrows = BB * LL;                             // 8192

    // Phase 1: 512 waves (16 rows each) -> 16384 threads -> 64 blocks
    proj_wmma_kernel<<<dim3((nrows / 16 * 32 + 255) / 256), 256, 0, stream>>>(
        et, table, wg, wsv, wd, bg, bs, bd, proj, nrows);

    // Phase 1b: hidden vector, B*L*64 threads
    hidden_kernel<<<dim3((BB * LL * 2 * DD) / 256), 256, 0, stream>>>(
        time, et, Wt, table, hidden);

    // Phase 2: grid (L/128, L/8, B) = (16, 256, 4), 256 threads/block
    pair_kernel<<<dim3(LL / 128, LL / 8, BB), 256, 0, stream>>>(
        time, proj, scores, tdout);
}